// CellList_40295383171536
// MI455X (gfx1250) — compile-verified
//
#include <hip/hip_runtime.h>
#include <hip/hip_bf16.h>

typedef __attribute__((ext_vector_type(2))) float v2f;
typedef __attribute__((ext_vector_type(8))) float v8f;

// Each wave computes one 16x16 tile of the NxN distance matrix.
//   d2[i][j] = r_i - 2*c_i.c_j + r_j   via V_WMMA_F32_16X16X4_F32:
//   A row i  = (x_i, y_i, z_i, r_i)         (16x4 f32)
//   B col j  = (-2x_j, -2y_j, -2z_j, 1)     (4x16 f32)
//   C        = r_j broadcast per column     (16x16 f32)
// Tiles strictly above the diagonal are pure zero-fill (output is poisoned by
// the harness, so every element must be written). The kernel is output-store
// bound (151 MB @ 23.3 TB/s ~ 6.5 us floor), so the epilogue is kept flat:
// raw v_sqrt_f32 computed unconditionally + v_cndmask select, no branches.
__global__ __launch_bounds__(256)
void celllist_dist_wmma(const int* __restrict__ species,
                        const float* __restrict__ coords,
                        const int* __restrict__ cutoff_p,
                        float* __restrict__ out,
                        int n, int nt) {
  const int lane = threadIdx.x & 31;
  const int wave = threadIdx.x >> 5;
  const int tile = blockIdx.x * 8 + wave;
  if (tile >= nt * nt) return;

  const int ti = tile / nt;          // tile row
  const int tj = tile - ti * nt;     // tile col

  const int half = lane >> 4;        // 0: lanes 0-15, 1: lanes 16-31
  const int l16  = lane & 15;
  const int row_base = ti * 16 + half * 8;   // C/D layout: +8 rows for hi half
  const int col = tj * 16 + l16;

  float* __restrict__ outp = out + (size_t)row_base * (size_t)n + col;

  if (tj > ti) {
    // Entire tile is i < j: masked out -> write zeros, no math.
    #pragma unroll
    for (int v = 0; v < 8; ++v)
      outp[(size_t)v * (size_t)n] = 0.0f;
    return;
  }

  // --- hoisted loads: latency hides under the WMMA ---------------------
  const float cut  = (float)cutoff_p[0];
  const float cut2 = cut * cut;

  const int i = ti * 16 + l16;       // row atom for this lane
  const int j = col;                 // col atom for this lane
  const float xi = coords[3 * i + 0];
  const float yi = coords[3 * i + 1];
  const float zi = coords[3 * i + 2];
  const float xj = coords[3 * j + 0];
  const float yj = coords[3 * j + 1];
  const float zj = coords[3 * j + 2];

  const bool vj = species[j] >= 0;
  bool vrow[8];
  #pragma unroll
  for (int v = 0; v < 8; ++v)
    vrow[v] = species[row_base + v] >= 0;   // L0/L2 resident (24 KB total)

  const float ri = xi * xi + yi * yi + zi * zi;
  const float rj = xj * xj + yj * yj + zj * zj;

  // A-matrix 16x4 f32 layout: lanes 0-15 -> {K0,K1}, lanes 16-31 -> {K2,K3}
  v2f a;
  a.x = half ? zi : xi;
  a.y = half ? ri : yi;
  // B-matrix 4x16 f32 layout mirrors A (row k striped across lanes)
  v2f b;
  b.x = half ? (-2.0f * zj) : (-2.0f * xj);
  b.y = half ? 1.0f         : (-2.0f * yj);
  // C-matrix: column n of tile gets +r_j in every row
  v8f c;
  #pragma unroll
  for (int v = 0; v < 8; ++v) c[v] = rj;

  // D = A x B + C  ->  full 16x16 tile of squared distances in one WMMA.
  v8f d2 = __builtin_amdgcn_wmma_f32_16x16x4_f32(
      /*neg_a=*/false, a, /*neg_b=*/false, b,
      /*c_mod=*/(short)0, c, /*reuse_a=*/false, /*reuse_b=*/false);

  // --- flat epilogue: no divergent branches --------------------------------
  #pragma unroll
  for (int v = 0; v < 8; ++v) {
    const int row = row_base + v;
    // clamp guards tiny negative q from the r-2ab+r cancellation; does not
    // change the cutoff test (negative q passes <= cut2 either way).
    const float q = fmaxf(d2[v], 0.0f);
    const float s = __builtin_amdgcn_sqrtf(q);     // raw v_sqrt_f32
    const bool keep = (row > col) & vrow[v] & vj & (q <= cut2);
    outp[(size_t)v * (size_t)n] = keep ? s : 0.0f; // v_cndmask + store
  }
}

extern "C" void kernel_launch(void* const* d_in, const int* in_sizes, int n_in,
                              void* d_out, int out_size, void* d_ws, size_t ws_size,
                              hipStream_t stream) {
  const int*   species = (const int*)d_in[0];    // [1, N] int
  const float* coords  = (const float*)d_in[1];  // [1, N, 3] f32
  const int*   cutoff  = (const int*)d_in[2];    // scalar int
  float* out = (float*)d_out;                    // [N, N] f32

  const int n  = in_sizes[0];      // 6144
  const int nt = n / 16;           // 384 tiles per dimension
  const int tiles = nt * nt;       // 147456 tiles, one wave each
  const int blocks = (tiles + 7) / 8;  // 8 waves (256 threads) per block

  celllist_dist_wmma<<<blocks, 256, 0, stream>>>(species, coords, cutoff,
                                                 out, n, nt);
}